// DeltaSynapse_64398739636541
// MI455X (gfx1250) — compile-verified
//
#include <hip/hip_runtime.h>

// DeltaSynapse: I[b,o] = sum_e signs[e,o]*W[e,o] * Xd[didx(e,o), b, e]
// Reformulated as a single fp32 GEMM  I = A @ Beff  with
//   M = 16 (batch), N = 2048 (post neurons), K = 16384 = e*8+d,
//   A[b, e*8+d]   = Xd[d,b,e]
//   Beff[e*8+d,o] = delaymap[d,e,o] * signs[e,o] * W[e,o]   (formed on the fly)
// Memory-bound (~161 MB traffic vs 1.07 GFLOP -> ~7us at 23.3 TB/s), so we
// keep full fp32 precision and use V_WMMA_F32_16X16X4_F32 (M=16 == batch).

typedef __attribute__((ext_vector_type(2))) float v2f;
typedef __attribute__((ext_vector_type(8))) float v8f;

constexpr int kD = 8;
constexpr int kB = 16;
constexpr int kN = 2048;

constexpr int WAVES_PER_WG = 16;           // 512 threads
constexpr int EPW          = kN / WAVES_PER_WG;  // 128 e-values per wave

__global__ __launch_bounds__(512)
void delta_synapse_kernel(const float* __restrict__ Xd,        // (D,B,N)
                          const float* __restrict__ delaymap,  // (D,N,N)
                          const float* __restrict__ W,         // (N,N)
                          const float* __restrict__ signs,     // (N,N)
                          float* __restrict__ out)             // (B,N)
{
    __shared__ float red[WAVES_PER_WG * 256];   // 16 wave-partial 16x16 tiles

    const int tid  = threadIdx.x;
    const int wave = tid >> 5;        // 0..15
    const int lane = tid & 31;
    const int half = lane >> 4;       // 0 / 1 : selects K rows per WMMA layout
    const int l16  = lane & 15;

    const int otile = blockIdx.x * 16;          // this WG's 16 output columns
    const int o     = otile + l16;              // lane's output column
    const int b     = l16;                      // lane's batch row (M)
    const int e0    = wave * EPW;               // wave's K slice (in e units)

    v8f acc = {};

    for (int e = e0; e < e0 + EPW; ++e) {
        // Effective weight for this (e, o): identical in both lane halves.
        const float weff =
            __builtin_nontemporal_load(&signs[(size_t)e * kN + o]) *
            __builtin_nontemporal_load(&W[(size_t)e * kN + o]);

        const float* dm_e = delaymap + (size_t)e * kN + o;   // + d*N*N per plane
        const float* xd_e = Xd + (size_t)b * kN + e;         // + d*B*N per plane

        // Stream-prefetch delaymap ~8 iterations ahead; cycle the 8 d-planes
        // so each plane gets prefetched once per 8 e's.
        __builtin_prefetch(dm_e + (size_t)(e & 7) * kN * kN + 8 * kN, 0, 0);

        // K-chunk 0: k = e*8 + {0,1,2,3}
        {
            const int dA = 2 * half;            // lanes 0-15 -> d0,d1 ; 16-31 -> d2,d3
            v2f a, bv;
            a.x  = xd_e[(size_t)(dA + 0) * kB * kN];
            a.y  = xd_e[(size_t)(dA + 1) * kB * kN];
            bv.x = __builtin_nontemporal_load(&dm_e[(size_t)(dA + 0) * kN * kN]) * weff;
            bv.y = __builtin_nontemporal_load(&dm_e[(size_t)(dA + 1) * kN * kN]) * weff;
            acc = __builtin_amdgcn_wmma_f32_16x16x4_f32(
                      false, a, false, bv, (short)0, acc, false, false);
        }
        // K-chunk 1: k = e*8 + {4,5,6,7}
        {
            const int dA = 4 + 2 * half;
            v2f a, bv;
            a.x  = xd_e[(size_t)(dA + 0) * kB * kN];
            a.y  = xd_e[(size_t)(dA + 1) * kB * kN];
            bv.x = __builtin_nontemporal_load(&dm_e[(size_t)(dA + 0) * kN * kN]) * weff;
            bv.y = __builtin_nontemporal_load(&dm_e[(size_t)(dA + 1) * kN * kN]) * weff;
            acc = __builtin_amdgcn_wmma_f32_16x16x4_f32(
                      false, a, false, bv, (short)0, acc, false, false);
        }
    }

    // Deterministic LDS serial reduction of the 16 wave partials.
    // All waves share the same C/D VGPR layout, so a flat elementwise sum is valid.
    #pragma unroll
    for (int r = 0; r < 8; ++r)
        red[wave * 256 + r * 32 + lane] = acc[r];
    __syncthreads();

    if (tid < 256) {
        float s = 0.0f;
        #pragma unroll
        for (int w = 0; w < WAVES_PER_WG; ++w)
            s += red[w * 256 + tid];
        // Flat idx = r*32 + lane maps to C/D layout:
        //   row  M = r + 8*(lane>>4)  (batch b)
        //   col  N = lane & 15        (output o within tile)
        const int r2 = tid >> 5;
        const int ln = tid & 31;
        const int ob = r2 + 8 * (ln >> 4);
        const int oo = otile + (ln & 15);
        out[(size_t)ob * kN + oo] = s;
    }
}

extern "C" void kernel_launch(void* const* d_in, const int* in_sizes, int n_in,
                              void* d_out, int out_size, void* d_ws, size_t ws_size,
                              hipStream_t stream) {
    const float* Xd       = (const float*)d_in[0];
    // d_in[1] = X (post spikes) — unused by the reference core path
    const float* delaymap = (const float*)d_in[2];
    const float* W        = (const float*)d_in[3];
    const float* signs    = (const float*)d_in[4];
    float* out = (float*)d_out;   // (B, N) fp32

    dim3 grid(kN / 16);   // 128 workgroups, one per 16-wide output tile
    dim3 block(512);      // 16 waves; each wave owns a K-slice of 128 e's
    hipLaunchKernelGGL(delta_synapse_kernel, grid, block, 0, stream,
                       Xd, delaymap, W, signs, out);
}